// LSTNet_25366076850398
// MI455X (gfx1250) — compile-verified
//
#include <hip/hip_runtime.h>
#include <hip/hip_bf16.h>

typedef float v2f __attribute__((ext_vector_type(2)));
typedef float v8f __attribute__((ext_vector_type(8)));

// ---------------------------------------------------------------------------
// Generic channel-first GEMM via V_WMMA_F32_16X16X4_F32.
//   OUT[b,o,n] = act( sum_c W[o,c]*IN[b,c,n] + bias[o] (+ RES[b,o,n]) )
// REQUIRES: Cin % 4 == 0, Cout % 16 == 0, N % 16 == 0 (callers pre-pad).
// One wave computes a 16(o) x 16(n) tile; 8 waves per block cover 128 n.
// A frag (16x4 f32): lane half selects K pair, vgpr selects element.
// B frag (4x16 f32): vgpr v -> K = v + (lane>>4)*2, N = lane&15.
// C/D (16x16 f32): vgpr r -> M = r + (lane>>4)*8, N = lane&15.
// ---------------------------------------------------------------------------
__global__ __launch_bounds__(256)
void gemm_cf_kernel(const float* __restrict__ W,    // (Cout, Cin), Cin%4==0
                    const float* __restrict__ bias, // (Cout)
                    const float* __restrict__ IN,   // (B, Cin, N)
                    float* __restrict__ OUT,        // (B, Cout, N)
                    const float* __restrict__ RES,  // (B, Cout, N) or nullptr
                    int Cin, int Cout, int N, int act)
{
    const int lane = threadIdx.x & 31;
    const int wv   = threadIdx.x >> 5;
    const int b    = blockIdx.z;
    const int o0   = blockIdx.y * 16;
    const int n0   = blockIdx.x * 128 + wv * 16;
    if (n0 >= N) return;

    const int m  = lane & 15;
    const int hk = (lane >> 4) * 2;   // 0 or 2
    const int o  = o0 + m;            // A-row for this lane
    const int n  = n0 + m;            // B-col for this lane

    // Unconditional streaming pointers (even float offset -> 8B-aligned v2f).
    const float* wp = W + (size_t)o * Cin + hk;
    const float* bp = IN + (size_t)b * Cin * N + (size_t)hk * N + n;
    const size_t bstep = (size_t)4 * N;

    v8f acc = {0.f, 0.f, 0.f, 0.f, 0.f, 0.f, 0.f, 0.f};

    #pragma unroll 2
    for (int c0 = 0; c0 < Cin; c0 += 4) {
        v2f a = *(const v2f*)wp;      // W[o][c0+hk], W[o][c0+hk+1]
        v2f bb;
        bb.x = bp[0];                 // IN[c0+hk][n]
        bb.y = bp[N];                 // IN[c0+hk+1][n]
        acc = __builtin_amdgcn_wmma_f32_16x16x4_f32(false, a, false, bb,
                                                    (short)0, acc, false, false);
        wp += 4;
        bp += bstep;
    }

    const int oh = o0 + (lane >> 4) * 8;
    const size_t outbase = (size_t)b * Cout * N;
    #pragma unroll
    for (int r = 0; r < 8; ++r) {
        const int oo = oh + r;
        float v = acc[r] + bias[oo];
        if (RES) v += RES[outbase + (size_t)oo * N + n];
        if (act == 1)      v = v > 0.f ? v : 0.f;          // relu
        else if (act == 2) v = v >= 0.f ? v : 0.2f * v;    // lrelu(0.2)
        OUT[outbase + (size_t)oo * N + n] = v;
    }
}

// ---------------------------------------------------------------------------
// Weight / bias zero-padding into workspace (ragged shapes -> WMMA-friendly)
// ---------------------------------------------------------------------------
__global__ void pad_weight_kernel(const float* __restrict__ src, float* __restrict__ dst,
                                  int Cout, int Cin, int CoutP, int CinP)
{
    const int t = blockIdx.x * 256 + threadIdx.x;
    if (t >= CoutP * CinP) return;
    const int o = t / CinP, c = t - o * CinP;
    dst[t] = (o < Cout && c < Cin) ? src[o * Cin + c] : 0.f;
}

__global__ void pad_bias_kernel(const float* __restrict__ src, float* __restrict__ dst,
                                int Cout, int CoutP)
{
    const int t = blockIdx.x * 256 + threadIdx.x;
    if (t >= CoutP) return;
    dst[t] = (t < Cout) ? src[t] : 0.f;
}

// ---------------------------------------------------------------------------
// Farthest point sampling: one block per batch, sequential NP iterations.
// ---------------------------------------------------------------------------
__global__ __launch_bounds__(1024)
void fps_kernel(const float* __restrict__ pc,   // (B,3,Npts)
                int* __restrict__ fps_idx,      // (B,NP)
                float* __restrict__ new_xyz,    // (B,NP,3)
                int Npts, int NP)
{
    const int b = blockIdx.x;
    const int tid = threadIdx.x;
    const float* px = pc + (size_t)b * 3 * Npts;
    const float* py = px + Npts;
    const float* pz = py + Npts;

    __shared__ float sval[1024];
    __shared__ int   sidx[1024];
    __shared__ int   s_far;
    if (tid == 0) s_far = 0;
    __syncthreads();

    float md0 = 1e10f, md1 = 1e10f;
    const int n0 = tid, n1 = tid + 1024;

    for (int it = 0; it < NP; ++it) {
        const int far = s_far;
        const float cx = px[far], cy = py[far], cz = pz[far];
        if (tid == 0) {
            fps_idx[(size_t)b * NP + it] = far;
            new_xyz[((size_t)b * NP + it) * 3 + 0] = cx;
            new_xyz[((size_t)b * NP + it) * 3 + 1] = cy;
            new_xyz[((size_t)b * NP + it) * 3 + 2] = cz;
        }
        {
            float dx = px[n0] - cx, dy = py[n0] - cy, dz = pz[n0] - cz;
            md0 = fminf(md0, dx * dx + dy * dy + dz * dz);
        }
        if (n1 < Npts) {
            float dx = px[n1] - cx, dy = py[n1] - cy, dz = pz[n1] - cz;
            md1 = fminf(md1, dx * dx + dy * dy + dz * dz);
        }
        float best = md0; int bi = n0;
        if (n1 < Npts && md1 > best) { best = md1; bi = n1; }
        sval[tid] = best; sidx[tid] = bi;
        __syncthreads();
        for (int s = 512; s > 0; s >>= 1) {
            if (tid < s) {
                float v2 = sval[tid + s]; int i2 = sidx[tid + s];
                if (v2 > sval[tid] || (v2 == sval[tid] && i2 < sidx[tid])) {
                    sval[tid] = v2; sidx[tid] = i2;
                }
            }
            __syncthreads();
        }
        if (tid == 0) s_far = sidx[0];
        __syncthreads();
    }
}

// ---------------------------------------------------------------------------
// KNN: one block per (b, query m); distances in LDS, K rounds of argmin.
// ---------------------------------------------------------------------------
__global__ __launch_bounds__(256)
void knn_kernel(const float* __restrict__ ref,    // (B,Nref,3) point-major
                const float* __restrict__ query,  // (B,M,3)
                int* __restrict__ out_idx,        // (B,M,K)
                int Nref, int M, int K)
{
    __shared__ float d[2048];
    __shared__ float rval[256];
    __shared__ int   ridx[256];
    const int b = blockIdx.y, m = blockIdx.x, tid = threadIdx.x;
    const float* rb = ref + (size_t)b * Nref * 3;
    const float* q  = query + ((size_t)b * M + m) * 3;
    const float qx = q[0], qy = q[1], qz = q[2];

    for (int n = tid; n < Nref; n += 256) {
        float dx = qx - rb[n * 3], dy = qy - rb[n * 3 + 1], dz = qz - rb[n * 3 + 2];
        d[n] = dx * dx + dy * dy + dz * dz;
    }
    __syncthreads();

    for (int k = 0; k < K; ++k) {
        float bv = 3.4e38f; int bi = 0;
        for (int n = tid; n < Nref; n += 256) {
            float v = d[n];
            if (v < bv) { bv = v; bi = n; }
        }
        rval[tid] = bv; ridx[tid] = bi;
        __syncthreads();
        for (int s = 128; s > 0; s >>= 1) {
            if (tid < s) {
                float v2 = rval[tid + s]; int i2 = ridx[tid + s];
                if (v2 < rval[tid] || (v2 == rval[tid] && i2 < ridx[tid])) {
                    rval[tid] = v2; ridx[tid] = i2;
                }
            }
            __syncthreads();
        }
        if (tid == 0) {
            out_idx[((size_t)b * M + m) * K + k] = ridx[0];
            d[ridx[0]] = 3.4e38f;
        }
        __syncthreads();
    }
}

// ---------------------------------------------------------------------------
// Small elementwise / gather / reduce kernels
// ---------------------------------------------------------------------------
__global__ void transpose_c3_kernel(const float* __restrict__ pc, float* __restrict__ xyz, int Npts)
{   // (B,3,Npts) -> (B,Npts,3)
    const int b = blockIdx.y;
    const int n = blockIdx.x * 256 + threadIdx.x;
    if (n >= Npts) return;
    for (int c = 0; c < 3; ++c)
        xyz[((size_t)b * Npts + n) * 3 + c] = pc[((size_t)b * 3 + c) * Npts + n];
}

__global__ void build_newpoints_kernel(const float* __restrict__ pc, const float* __restrict__ new_xyz,
                                       const int* __restrict__ idx, float* __restrict__ np_,
                                       int Npts, int M, int S)
{   // np_ (B,8,M*S): ch 0..2 = grouped - center, ch 3..5 = grouped, ch 6..7 = 0 (pad)
    const int b = blockIdx.z;
    const int t = blockIdx.x * 256 + threadIdx.x;
    if (t >= M * S) return;
    const int m = t / S, s = t - m * S;
    const int id = idx[((size_t)b * M + m) * S + s];
    const size_t MS = (size_t)M * S;
    for (int c = 0; c < 3; ++c) {
        float g   = pc[((size_t)b * 3 + c) * Npts + id];
        float cen = new_xyz[((size_t)b * M + m) * 3 + c];
        np_[((size_t)b * 8 + c)     * MS + t] = g - cen;
        np_[((size_t)b * 8 + c + 3) * MS + t] = g;
    }
    np_[((size_t)b * 8 + 6) * MS + t] = 0.f;
    np_[((size_t)b * 8 + 7) * MS + t] = 0.f;
}

__global__ void maxpool_lastdim_kernel(const float* __restrict__ in, float* __restrict__ out,
                                       int C, int M, int S)
{   // (B,C,M*S) -> (B,C,M)
    const int b = blockIdx.z;
    const int t = blockIdx.x * 256 + threadIdx.x;
    if (t >= C * M) return;
    const int c = t / M, m = t - c * M;
    const float* p = in + ((size_t)b * C + c) * ((size_t)M * S) + (size_t)m * S;
    float v = p[0];
    for (int s = 1; s < S; ++s) v = fmaxf(v, p[s]);
    out[((size_t)b * C + c) * M + m] = v;
}

__global__ void group_feat_kernel(const float* __restrict__ feat, // (B,C,M)
                                  const int* __restrict__ idx,    // (B,M,S)
                                  float* __restrict__ out,        // (B,C,M*S)
                                  int C, int M, int S)
{
    const int b = blockIdx.z;
    const int t = blockIdx.x * 256 + threadIdx.x;
    const int tot = C * M * S;
    if (t >= tot) return;
    const int c = t / (M * S); const int r = t - c * (M * S);
    const int m = r / S; const int s = r - m * S;
    const int id = idx[((size_t)b * M + m) * S + s];
    out[((size_t)b * C + c) * ((size_t)M * S) + r] = feat[((size_t)b * C + c) * M + id];
}

__global__ void pos_rel_kernel(const float* __restrict__ new_xyz, const int* __restrict__ idx,
                               float* __restrict__ out, int M, int S)
{   // out (B,4,M*S): ch 0..2 = pos[m]-pos[idx[m,s]], ch 3 = 0 (pad)
    const int b = blockIdx.z;
    const int t = blockIdx.x * 256 + threadIdx.x;
    if (t >= M * S) return;
    const int m = t / S, s = t - m * S;
    const int id = idx[((size_t)b * M + m) * S + s];
    const size_t MS = (size_t)M * S;
    for (int c = 0; c < 3; ++c)
        out[((size_t)b * 4 + c) * MS + t] =
            new_xyz[((size_t)b * M + m) * 3 + c] - new_xyz[((size_t)b * M + id) * 3 + c];
    out[((size_t)b * 4 + 3) * MS + t] = 0.f;
}

__global__ void attn_in_kernel(const float* __restrict__ query, // (B,C,M)
                               const float* __restrict__ pe,    // (B,C,M*S)
                               float* __restrict__ kg,          // in: key_g / out: attn input
                               int C, int M, int S)
{
    const int b = blockIdx.z;
    const int t = blockIdx.x * 256 + threadIdx.x;
    const int tot = C * M * S;
    if (t >= tot) return;
    const int c = t / (M * S); const int r = t - c * (M * S); const int m = r / S;
    const size_t o = (size_t)b * tot + t;
    kg[o] = query[((size_t)b * C + c) * M + m] - kg[o] + pe[o];
}

__global__ void softmax_s_kernel(float* __restrict__ a, int C, int M, int S)
{   // softmax over trailing S
    const int b = blockIdx.z;
    const int t = blockIdx.x * 256 + threadIdx.x;
    if (t >= C * M) return;
    float* p = a + ((size_t)b * C * M + t) * S;
    float mx = p[0];
    for (int s = 1; s < S; ++s) mx = fmaxf(mx, p[s]);
    float sum = 0.f;
    for (int s = 0; s < S; ++s) { float e = expf(p[s] - mx); p[s] = e; sum += e; }
    const float inv = 1.f / sum;
    for (int s = 0; s < S; ++s) p[s] *= inv;
}

__global__ void agg_kernel(const float* __restrict__ attn, const float* __restrict__ value,
                           const float* __restrict__ pe, float* __restrict__ agg,
                           int C, int M, int S)
{   // agg[b,c,m] = sum_s attn * (value[b,c,m] + pe)
    const int b = blockIdx.z;
    const int t = blockIdx.x * 256 + threadIdx.x;
    if (t >= C * M) return;
    const float* pa = attn + ((size_t)b * C * M + t) * S;
    const float* pp = pe   + ((size_t)b * C * M + t) * S;
    const float v = value[(size_t)b * C * M + t];
    float sum = 0.f;
    for (int s = 0; s < S; ++s) sum += pa[s] * (v + pp[s]);
    agg[(size_t)b * C * M + t] = sum;
}

__global__ void gmax_kernel(const float* __restrict__ feat, float* __restrict__ gf, int C, int N)
{   // gf[b,c] = max_n feat[b,c,n]
    const int b = blockIdx.y;
    const int c = blockIdx.x * 256 + threadIdx.x;
    if (c >= C) return;
    const float* p = feat + ((size_t)b * C + c) * N;
    float v = p[0];
    for (int n = 1; n < N; ++n) v = fmaxf(v, p[n]);
    gf[(size_t)b * C + c] = v;
}

__global__ void cat_gf_kernel(const float* __restrict__ feat, const float* __restrict__ gf,
                              float* __restrict__ out, int C, int N)
{   // out (B,2C,N): first C = feat, last C = gf broadcast
    const int b = blockIdx.z;
    const int t = blockIdx.x * 256 + threadIdx.x;
    if (t >= 2 * C * N) return;
    const int c2 = t / N, n = t - c2 * N;
    float v = (c2 < C) ? feat[((size_t)b * C + c2) * N + n] : gf[(size_t)b * C + (c2 - C)];
    out[((size_t)b * 2 * C + c2) * N + n] = v;
}

__global__ void finalize_kernel(const float* __restrict__ ret,     // (B,RC,M) channel-first
                                const float* __restrict__ new_xyz, // (B,M,3)
                                float* __restrict__ out_coarse,    // (B,3,2M)
                                float* __restrict__ out_sym,       // (B,3,M)
                                int M, int RC)
{
    const int b = blockIdx.y;
    const int n = blockIdx.x * 256 + threadIdx.x;
    if (n >= M) return;
    const float* rb = ret + (size_t)b * RC * M;
    const float kp[3] = { new_xyz[((size_t)b * M + n) * 3 + 0],
                          new_xyz[((size_t)b * M + n) * 3 + 1],
                          new_xyz[((size_t)b * M + n) * 3 + 2] };
    for (int dd = 0; dd < 3; ++dd) {
        float s = kp[0] * rb[(0 * 3 + dd) * (size_t)M + n]
                + kp[1] * rb[(1 * 3 + dd) * (size_t)M + n]
                + kp[2] * rb[(2 * 3 + dd) * (size_t)M + n]
                + rb[(9 + dd) * (size_t)M + n];
        out_sym[((size_t)b * 3 + dd) * M + n] = s;
        out_coarse[((size_t)b * 3 + dd) * 2 * M + n]     = s;      // sym half
        out_coarse[((size_t)b * 3 + dd) * 2 * M + M + n] = kp[dd]; // keypoints half
    }
}

__global__ void copy_kernel(const float* __restrict__ src, float* __restrict__ dst, size_t n)
{
    size_t i = (size_t)blockIdx.x * 256 + threadIdx.x;
    if (i < n) dst[i] = src[i];
}

// ---------------------------------------------------------------------------
// Host orchestration
// ---------------------------------------------------------------------------
static inline void gemm_cf(const float* W, const float* bias, const float* IN, float* OUT,
                           const float* RES, int B_, int Cin, int Cout, int N, int act,
                           hipStream_t stream)
{
    dim3 grid((N + 127) / 128, Cout / 16, B_);
    gemm_cf_kernel<<<grid, 256, 0, stream>>>(W, bias, IN, OUT, RES, Cin, Cout, N, act);
}

extern "C" void kernel_launch(void* const* d_in, const int* in_sizes, int n_in,
                              void* d_out, int out_size, void* d_ws, size_t ws_size,
                              hipStream_t stream)
{
    (void)in_sizes; (void)n_in; (void)out_size; (void)ws_size;
    const int B = 32, NPTS = 2048, NP = 512, S = 16;
    const int MS = NP * S; // 8192

    const float* pc = (const float*)d_in[0];
    const float *sa_w1 = (const float*)d_in[1],  *sa_b1 = (const float*)d_in[2];
    const float *sa_w2 = (const float*)d_in[3],  *sa_b2 = (const float*)d_in[4];
    const float *t_st_w = (const float*)d_in[5], *t_st_b = (const float*)d_in[6];
    const float *t_q_w = (const float*)d_in[7],  *t_q_b = (const float*)d_in[8];
    const float *t_k_w = (const float*)d_in[9],  *t_k_b = (const float*)d_in[10];
    const float *t_v_w = (const float*)d_in[11], *t_v_b = (const float*)d_in[12];
    const float *t_p1_w = (const float*)d_in[13], *t_p1_b = (const float*)d_in[14];
    const float *t_p2_w = (const float*)d_in[15], *t_p2_b = (const float*)d_in[16];
    const float *t_a1_w = (const float*)d_in[17], *t_a1_b = (const float*)d_in[18];
    const float *t_a2_w = (const float*)d_in[19], *t_a2_b = (const float*)d_in[20];
    const float *t_e_w = (const float*)d_in[21], *t_e_b = (const float*)d_in[22];
    const float *e1_w = (const float*)d_in[23], *e1_b = (const float*)d_in[24];
    const float *e2_w = (const float*)d_in[25], *e2_b = (const float*)d_in[26];
    const float *m1_w = (const float*)d_in[27], *m1_b = (const float*)d_in[28];
    const float *m2_w = (const float*)d_in[29], *m2_b = (const float*)d_in[30];
    const float *m3_w = (const float*)d_in[31], *m3_b = (const float*)d_in[32];
    const float *m4_w = (const float*)d_in[33], *m4_b = (const float*)d_in[34];

    // ---- workspace carve (floats, 16B-aligned chunks), ~330 MB peak ----
    float* ws = (float*)d_ws;
    size_t off = 0;
    auto alloc = [&](size_t nf) { nf = (nf + 3) & ~(size_t)3; float* p = ws + off; off += nf; return p; };
    float* xyz      = alloc((size_t)B * NPTS * 3);
    float* new_xyz  = alloc((size_t)B * NP * 3);
    int*   fps_idx  = (int*)alloc((size_t)B * NP);
    int*   idx1     = (int*)alloc((size_t)B * NP * S);
    int*   idx_knn  = (int*)alloc((size_t)B * NP * S);
    float* keyfeat  = alloc((size_t)B * 128 * NP);
    float* x_feat   = alloc((size_t)B * 64 * NP);
    float* q_feat   = alloc((size_t)B * 64 * NP);
    float* k_feat   = alloc((size_t)B * 64 * NP);
    float* v_feat   = alloc((size_t)B * 64 * NP);
    float* agg_b    = alloc((size_t)B * 64 * NP);
    float* keyfeat2 = alloc((size_t)B * 128 * NP);
    float* gf       = alloc((size_t)B * 512);
    float* wp_sa1   = alloc(64 * 8);     // sa_w1 padded (64,6)->(64,8)
    float* wp_p1    = alloc(64 * 4);     // t_pos_w1 padded (64,3)->(64,4)
    float* wp_m4    = alloc(16 * 128);   // mlp_w4 padded (12,128)->(16,128)
    float* bp_m4    = alloc(16);         // mlp_b4 padded (12)->(16)
    float* np_buf   = alloc((size_t)B * 8 * MS);    // new_points(8ch) -> pos_rel(4ch) -> ret(16ch)
    float* ah_buf   = alloc((size_t)256 * MS);      // attn hidden (1 batch) -> mlp_h3
    float* bigA     = alloc((size_t)B * 64 * MS);   // h1 -> pe_tmp -> feat1024
    float* bigB     = alloc((size_t)B * 128 * MS);  // h2 -> [pe | attn] -> [exp_h | feat]
    float* bigC     = alloc((size_t)B * 64 * MS);   // key_g/attn_in -> [mlp_h1 | mlp_h2]

    float* pe      = bigB;
    float* attn    = bigB + (size_t)B * 64 * MS;
    float* exp_h   = bigB;
    float* feat    = bigB + (size_t)B * 256 * NP;
    float* feat1k  = bigA;
    float* mlp_h1  = bigC;
    float* mlp_h2  = bigC + (size_t)B * 512 * NP;
    float* mlp_h3  = ah_buf;
    float* ret     = np_buf;                         // (B,16,NP)

    float* out_coarse = (float*)d_out;                       // (B,3,1024)
    float* out_sym    = out_coarse + (size_t)B * 3 * 2 * NP; // (B,3,512)
    float* out_kf     = out_sym + (size_t)B * 3 * NP;        // (B,128,512)

    // ---- weight padding (tiny, once per launch) ----
    pad_weight_kernel<<<(64 * 8 + 255) / 256, 256, 0, stream>>>(sa_w1, wp_sa1, 64, 6, 64, 8);
    pad_weight_kernel<<<(64 * 4 + 255) / 256, 256, 0, stream>>>(t_p1_w, wp_p1, 64, 3, 64, 4);
    pad_weight_kernel<<<(16 * 128 + 255) / 256, 256, 0, stream>>>(m4_w, wp_m4, 12, 128, 16, 128);
    pad_bias_kernel<<<1, 256, 0, stream>>>(m4_b, bp_m4, 12, 16);

    // ---- pipeline ----
    transpose_c3_kernel<<<dim3((NPTS + 255) / 256, B), 256, 0, stream>>>(pc, xyz, NPTS);
    fps_kernel<<<B, 1024, 0, stream>>>(pc, fps_idx, new_xyz, NPTS, NP);
    knn_kernel<<<dim3(NP, B), 256, 0, stream>>>(xyz, new_xyz, idx1, NPTS, NP, S);
    build_newpoints_kernel<<<dim3((MS + 255) / 256, 1, B), 256, 0, stream>>>(
        pc, new_xyz, idx1, np_buf, NPTS, NP, S);

    gemm_cf(wp_sa1, sa_b1, np_buf, bigA, nullptr, B, 8, 64, MS, 1, stream);   // h1 = relu
    gemm_cf(sa_w2, sa_b2, bigA, bigB, nullptr, B, 64, 128, MS, 0, stream);    // h2
    maxpool_lastdim_kernel<<<dim3((128 * NP + 255) / 256, 1, B), 256, 0, stream>>>(
        bigB, keyfeat, 128, NP, S);                                           // keyfeatures

    gemm_cf(t_st_w, t_st_b, keyfeat, x_feat, nullptr, B, 128, 64, NP, 0, stream); // x
    knn_kernel<<<dim3(NP, B), 256, 0, stream>>>(new_xyz, new_xyz, idx_knn, NP, NP, S);
    gemm_cf(t_k_w, t_k_b, x_feat, k_feat, nullptr, B, 64, 64, NP, 0, stream);
    gemm_cf(t_v_w, t_v_b, x_feat, v_feat, nullptr, B, 64, 64, NP, 0, stream);
    gemm_cf(t_q_w, t_q_b, x_feat, q_feat, nullptr, B, 64, 64, NP, 0, stream);

    group_feat_kernel<<<dim3((64 * MS + 255) / 256, 1, B), 256, 0, stream>>>(
        k_feat, idx_knn, bigC, 64, NP, S);                                    // key_g
    pos_rel_kernel<<<dim3((MS + 255) / 256, 1, B), 256, 0, stream>>>(
        new_xyz, idx_knn, np_buf, NP, S);                                     // pos_rel (4ch)
    gemm_cf(wp_p1, t_p1_b, np_buf, bigA, nullptr, B, 4, 64, MS, 1, stream);   // pe hidden
    gemm_cf(t_p2_w, t_p2_b, bigA, pe, nullptr, B, 64, 64, MS, 0, stream);     // pe
    attn_in_kernel<<<dim3((64 * MS + 255) / 256, 1, B), 256, 0, stream>>>(
        q_feat, pe, bigC, 64, NP, S);                                         // qk_rel + pe

    for (int b = 0; b < B; ++b) {                                             // attn MLP per batch
        const float* ain = bigC + (size_t)b * 64 * MS;
        float* aout = attn + (size_t)b * 64 * MS;
        gemm_cf(t_a1_w, t_a1_b, ain, ah_buf, nullptr, 1, 64, 256, MS, 1, stream);
        gemm_cf(t_a2_w, t_a2_b, ah_buf, aout, nullptr, 1, 256, 64, MS, 0, stream);
    }
    softmax_s_kernel<<<dim3((64 * NP + 255) / 256, 1, B), 256, 0, stream>>>(attn, 64, NP, S);
    agg_kernel<<<dim3((64 * NP + 255) / 256, 1, B), 256, 0, stream>>>(
        attn, v_feat, pe, agg_b, 64, NP, S);

    gemm_cf(t_e_w, t_e_b, agg_b, keyfeat2, keyfeat, B, 64, 128, NP, 0, stream); // + identity
    gemm_cf(e1_w, e1_b, keyfeat2, exp_h, nullptr, B, 128, 256, NP, 1, stream);
    gemm_cf(e2_w, e2_b, exp_h, feat, nullptr, B, 256, 512, NP, 0, stream);

    gmax_kernel<<<dim3((512 + 255) / 256, B), 256, 0, stream>>>(feat, gf, 512, NP);
    cat_gf_kernel<<<dim3((1024 * NP + 255) / 256, 1, B), 256, 0, stream>>>(
        feat, gf, feat1k, 512, NP);

    gemm_cf(m1_w, m1_b, feat1k, mlp_h1, nullptr, B, 1024, 512, NP, 2, stream);
    gemm_cf(m2_w, m2_b, mlp_h1, mlp_h2, nullptr, B, 512, 256, NP, 2, stream);
    gemm_cf(m3_w, m3_b, mlp_h2, mlp_h3, nullptr, B, 256, 128, NP, 2, stream);
    gemm_cf(wp_m4, bp_m4, mlp_h3, ret, nullptr, B, 128, 16, NP, 0, stream);   // ret (B,16,NP)

    finalize_kernel<<<dim3((NP + 255) / 256, B), 256, 0, stream>>>(
        ret, new_xyz, out_coarse, out_sym, NP, 16);
    const size_t nkf = (size_t)B * 128 * NP;
    copy_kernel<<<(unsigned)((nkf + 255) / 256), 256, 0, stream>>>(keyfeat2, out_kf, nkf);
}